// GCN_47158740910498
// MI455X (gfx1250) — compile-verified
//
#include <hip/hip_runtime.h>
#include <hip/hip_bf16.h>

// ---------------------------------------------------------------------------
// 2-layer GCN for MI455X (gfx1250, wave32).
//   layer(x): h = x@W ; out[dst] += h[src]*dinv[src]*dinv[dst] (+self loop) ; +b
//
// GEMM: V_WMMA_F32_16X16X4_F32, templated on (K, NOUT) so the fully unrolled
// K-loop uses immediate-offset addressing; A-tile staged in LDS (shared by all
// 8 waves of the block). Aggregation (the true bandwidth bottleneck: ~5 GB of
// edge traffic vs ~13 GFLOP of GEMM) uses float4 gathers + fp32 global atomics;
// h1 (102 MB) is L2-resident (192 MB).
// ---------------------------------------------------------------------------

typedef float v2f __attribute__((ext_vector_type(2)));
typedef float v8f __attribute__((ext_vector_type(8)));

// ---- degree / normalization ------------------------------------------------
__global__ void gcn_deg_init(float* __restrict__ deg, int n) {
    int i = blockIdx.x * blockDim.x + threadIdx.x;
    if (i < n) deg[i] = 1.0f;  // self-loop contributes 1 to every node
}

__global__ void gcn_deg_count(float* __restrict__ deg,
                              const int* __restrict__ dst, int e) {
    int i = blockIdx.x * blockDim.x + threadIdx.x;
    if (i < e) atomicAdd(&deg[dst[i]], 1.0f);
}

__global__ void gcn_dinv(float* __restrict__ deg, int n) {
    int i = blockIdx.x * blockDim.x + threadIdx.x;
    if (i < n) {
        float d = deg[i];
        deg[i] = (d > 0.0f) ? rsqrtf(d) : 0.0f;
    }
}

// ---- fp32 WMMA GEMM: C[M,NOUT] = A[M,K] @ W[K,NOUT], K/NOUT compile-time ---
// Block = 256 threads = 8 waves. Wave w owns the 16x16 tile at
// (blockIdx.x*16, (blockIdx.y*8+w)*16).
//
// ISA 16x4 f32 A layout (symmetric 4x16 B layout):
//   lanes 0-15 : row/col = lane,    VGPR0 = K+0, VGPR1 = K+1
//   lanes 16-31: row/col = lane-16, VGPR0 = K+2, VGPR1 = K+3
template <int K, int NOUT>
__global__ void gcn_gemm_wmma_t(const float* __restrict__ A,
                                const float* __restrict__ W,
                                float* __restrict__ C,
                                int M) {
    __shared__ float As[16 * K];            // 8/16 KB of the 320 KB WGP LDS

    const int tid   = threadIdx.x;
    const int lane  = tid & 31;
    const int lhalf = lane & 15;
    const int koff  = (lane >> 4) << 1;     // 0 for lanes 0-15, 2 for 16-31
    const int row0  = blockIdx.x * 16;
    const int col0  = (blockIdx.y * 8 + (tid >> 5)) * 16;

    // --- cooperative stage of the 16xK A-tile (float4, clamped rows) ---
#pragma unroll
    for (int i = tid * 4; i < 16 * K; i += 256 * 4) {
        int r  = i / K;
        int c  = i - r * K;
        int gr = row0 + r;
        if (gr >= M) gr = M - 1;            // clamp loads; stores are guarded
        *(float4*)(As + i) = *(const float4*)(A + (size_t)gr * K + c);
    }
    __syncthreads();

    const float* __restrict__ as = As + lhalf * K + koff;
    const float* __restrict__ wp = W + (size_t)koff * NOUT + col0 + lhalf;

    v8f acc = {};
#pragma unroll
    for (int k = 0; k < K; k += 4) {
        v2f av = *(const v2f*)(as + k);                 // ds_load_b64, imm off
        v2f bv;
        bv.x = wp[k * NOUT];                            // imm-offset b32 loads
        bv.y = wp[(k + 1) * NOUT];
        acc = __builtin_amdgcn_wmma_f32_16x16x4_f32(
            /*neg_a=*/false, av, /*neg_b=*/false, bv,
            /*c_mod=*/(short)0, acc, /*reuse_a=*/false, /*reuse_b=*/false);
    }

    // C/D layout: VGPR i -> M = i (+8 for upper half-wave), N = lane&15
    const int col = col0 + lhalf;
    const int rb  = row0 + ((lane >> 4) << 3);
#pragma unroll
    for (int i = 0; i < 8; ++i) {
        const int r = rb + i;
        if (r < M) C[(size_t)r * NOUT + col] = acc[i];
    }
}

// Dynamic-shape fallback (same algorithm, runtime K/Nout).
__global__ void gcn_gemm_wmma_dyn(const float* __restrict__ A,
                                  const float* __restrict__ W,
                                  float* __restrict__ C,
                                  int M, int K, int Nout) {
    const int lane  = threadIdx.x & 31;
    const int wave  = threadIdx.x >> 5;
    const int row0  = blockIdx.x * 16;
    const int col0  = (blockIdx.y * 8 + wave) * 16;
    const int lhalf = lane & 15;
    const int koff  = (lane >> 4) << 1;

    int arow = row0 + lhalf;
    if (arow >= M) arow = M - 1;
    const float* __restrict__ aptr = A + (size_t)arow * K + koff;
    const float* __restrict__ wptr = W + (size_t)koff * Nout + col0 + lhalf;
    const size_t wstep = (size_t)4 * Nout;

    v8f acc = {};
    for (int k = 0; k < K; k += 4) {
        v2f av; av.x = aptr[0];
                av.y = aptr[1];
        v2f bv; bv.x = wptr[0];
                bv.y = wptr[Nout];
        acc = __builtin_amdgcn_wmma_f32_16x16x4_f32(
            false, av, false, bv, (short)0, acc, false, false);
        aptr += 4;
        wptr += wstep;
    }

    const int col = col0 + lhalf;
    const int rb  = row0 + ((lane >> 4) << 3);
#pragma unroll
    for (int i = 0; i < 8; ++i) {
        const int r = rb + i;
        if (r < M) C[(size_t)r * Nout + col] = acc[i];
    }
}

// ---- aggregation -----------------------------------------------------------
// out[i,f] = h[i,f]*dinv[i]^2 + bias[f]  (self-loop + bias; bias is added once
// per node after the segment-sum, so folding it here is exact)
__global__ void gcn_agg_init(const float* __restrict__ h,
                             const float* __restrict__ dinv,
                             const float* __restrict__ bias,
                             float* __restrict__ out,
                             int n, int F4 /* = F/4 */) {
    long long i4 = (long long)blockIdx.x * blockDim.x + threadIdx.x;
    if (i4 < (long long)n * F4) {
        int node = (int)(i4 / F4);
        int f4   = (int)(i4 - (long long)node * F4);
        float di = dinv[node];
        float s  = di * di;
        float4 hv = ((const float4*)h)[i4];
        float4 bv = ((const float4*)bias)[f4];
        float4 o;
        o.x = hv.x * s + bv.x;
        o.y = hv.y * s + bv.y;
        o.z = hv.z * s + bv.z;
        o.w = hv.w * s + bv.w;
        ((float4*)out)[i4] = o;
    }
}

// One wave per edge: lanes stride the feature dim in float4s -> b128 gathers
// of h[src,*], fp32 global atomics into out[dst,*].
__global__ void gcn_agg_edges(const float* __restrict__ h,
                              const float* __restrict__ dinv,
                              const int* __restrict__ src,
                              const int* __restrict__ dst,
                              float* __restrict__ out,
                              int F4 /* = F/4 */, int e) {
    const int wavesPerBlock = blockDim.x >> 5;
    long long ei = (long long)blockIdx.x * wavesPerBlock + (threadIdx.x >> 5);
    if (ei >= e) return;
    const int lane = threadIdx.x & 31;
    const int s = src[ei];
    const int d = dst[ei];
    const float norm = dinv[s] * dinv[d];
    const float4* __restrict__ hs = (const float4*)(h + (size_t)s * F4 * 4);
    float* __restrict__ od = out + (size_t)d * F4 * 4;
    for (int f = lane; f < F4; f += 32) {
        float4 v = hs[f];
        int b = f * 4;
        atomicAdd(od + b + 0, v.x * norm);
        atomicAdd(od + b + 1, v.y * norm);
        atomicAdd(od + b + 2, v.z * norm);
        atomicAdd(od + b + 3, v.w * norm);
    }
}

__global__ void gcn_relu(float* __restrict__ x, long long n4) {
    long long i = (long long)blockIdx.x * blockDim.x + threadIdx.x;
    if (i < n4) {
        float4 v = ((const float4*)x)[i];
        v.x = v.x > 0.0f ? v.x : 0.0f;
        v.y = v.y > 0.0f ? v.y : 0.0f;
        v.z = v.z > 0.0f ? v.z : 0.0f;
        v.w = v.w > 0.0f ? v.w : 0.0f;
        ((float4*)x)[i] = v;
    }
}

// ---------------------------------------------------------------------------
extern "C" void kernel_launch(void* const* d_in, const int* in_sizes, int n_in,
                              void* d_out, int out_size, void* d_ws, size_t ws_size,
                              hipStream_t stream) {
    const float* x  = (const float*)d_in[0];
    const int*   ei = (const int*)d_in[1];
    const float* W1 = (const float*)d_in[2];
    const float* b1 = (const float*)d_in[3];
    const float* W2 = (const float*)d_in[4];
    const float* b2 = (const float*)d_in[5];
    float* out = (float*)d_out;

    const int HID = in_sizes[3];                 // 256
    const int OUT = in_sizes[5];                 // 128
    const int IN  = in_sizes[2] / HID;           // 128
    const int N   = in_sizes[0] / IN;            // 100000
    const int E   = in_sizes[1] / 2;             // 1600000

    const int* src = ei;
    const int* dst = ei + E;

    // workspace: dinv[N] | bufA[N*HID] | bufB[N*HID]
    float* dinv = (float*)d_ws;
    float* bufA = dinv + N;
    float* bufB = bufA + (size_t)N * HID;

    const int T = 256;

    auto launch_gemm = [&](const float* Am, const float* Wm, float* Cm,
                           int M, int Kd, int Nd) {
        dim3 grid((M + 15) / 16, Nd / 128);
        if (Kd == 128 && Nd == 256)
            gcn_gemm_wmma_t<128, 256><<<grid, T, 0, stream>>>(Am, Wm, Cm, M);
        else if (Kd == 256 && Nd == 128)
            gcn_gemm_wmma_t<256, 128><<<grid, T, 0, stream>>>(Am, Wm, Cm, M);
        else
            gcn_gemm_wmma_dyn<<<grid, T, 0, stream>>>(Am, Wm, Cm, M, Kd, Nd);
    };

    // --- symmetric normalization: deg -> dinv (shared by both layers) ---
    gcn_deg_init <<<(N + T - 1) / T, T, 0, stream>>>(dinv, N);
    gcn_deg_count<<<(E + T - 1) / T, T, 0, stream>>>(dinv, dst, E);
    gcn_dinv     <<<(N + T - 1) / T, T, 0, stream>>>(dinv, N);

    // --- layer 1: h1 = x @ W1 ; agg ; +b1 ; relu ---
    launch_gemm(x, W1, bufA, N, IN, HID);
    {
        long long tot4 = (long long)N * (HID / 4);
        gcn_agg_init <<<(unsigned)((tot4 + T - 1) / T), T, 0, stream>>>(bufA, dinv, b1, bufB, N, HID / 4);
        gcn_agg_edges<<<(E + 7) / 8, T, 0, stream>>>(bufA, dinv, src, dst, bufB, HID / 4, E);
        gcn_relu     <<<(unsigned)((tot4 + T - 1) / T), T, 0, stream>>>(bufB, tot4);
    }

    // --- layer 2: h2 = relu1 @ W2 ; agg ; +b2 -> d_out ---
    launch_gemm(bufB, W2, bufA, N, HID, OUT);
    {
        long long tot4 = (long long)N * (OUT / 4);
        gcn_agg_init <<<(unsigned)((tot4 + T - 1) / T), T, 0, stream>>>(bufA, dinv, b2, out, N, OUT / 4);
        gcn_agg_edges<<<(E + 7) / 8, T, 0, stream>>>(bufA, dinv, src, dst, out, OUT / 4, E);
    }
}